// FFF_43147241456089
// MI455X (gfx1250) — compile-verified
//
#include <hip/hip_runtime.h>

#define DEPTH 11
#define D 1024
#define H 32
#define O 1024
#define BATCH 4096
#define N_NODES 2047

typedef __attribute__((ext_vector_type(2))) float v2f;
typedef __attribute__((ext_vector_type(8))) float v8f;

// ---------------------------------------------------------------------------
// Kernel 1: hard tree descent. One wave32 per sample.
// x kept in 32 VGPRs (8 x float4); 11 gathered 1024-dot products with
// butterfly wave reduction. Node weights (8 MB total, top levels shared by
// all samples) live in L2.
// ---------------------------------------------------------------------------
__global__ __launch_bounds__(256) void fff_descent(
    const float* __restrict__ x, const float* __restrict__ nw,
    const float* __restrict__ nb, int* __restrict__ leaf_idx)
{
  const int tid    = blockIdx.x * 256 + threadIdx.x;
  const int sample = tid >> 5;
  const int lane   = threadIdx.x & 31;

  const float4* xs = (const float4*)(x + (size_t)sample * D);
  float4 xr[8];
#pragma unroll
  for (int j = 0; j < 8; ++j) xr[j] = xs[j * 32 + lane];

  int node = 0;
#pragma unroll
  for (int i = 0; i < DEPTH; ++i) {
    const float4* w = (const float4*)(nw + (size_t)node * D);
    float s = 0.f;
#pragma unroll
    for (int j = 0; j < 8; ++j) {
      float4 wv = w[j * 32 + lane];
      s += xr[j].x * wv.x + xr[j].y * wv.y + xr[j].z * wv.z + xr[j].w * wv.w;
    }
#pragma unroll
    for (int off = 16; off > 0; off >>= 1) s += __shfl_xor(s, off, 32);
    s += nb[node];
    node = 2 * node + 1 + (s >= 0.f ? 1 : 0);  // == (n-plat)*2+c+next_plat
  }
  if (lane == 0) leaf_idx[sample] = node - N_NODES;
}

// ---------------------------------------------------------------------------
// Kernel 2: per-sample leaf MLP via V_WMMA_F32_16X16X4_F32 (wave32).
// Bandwidth-dominant gathered weights go in the fully-utilized B operand
// (natural row-major K x N layout: n = lane&15, K-half = lane>>4, VGPR0/1
// stripe K within the half). Per-sample vector broadcast across A's 16 rows
// -> all C rows identical -> C[0] holds the output tile directly.
// ---------------------------------------------------------------------------
__global__ __launch_bounds__(256) void fff_leaf_mlp(
    const float* __restrict__ x,
    const float* __restrict__ w1s, const float* __restrict__ b1s,
    const float* __restrict__ w2s, const float* __restrict__ b2s,
    const int* __restrict__ leaf_idx, float* __restrict__ out)
{
  __shared__ float hsm[8 * H];                 // 8 waves/block * 32 floats
  const int wid    = threadIdx.x >> 5;
  const int lane   = threadIdx.x & 31;
  const int sample = blockIdx.x * 8 + wid;
  const int leaf   = leaf_idx[sample];
  const int nsub   = lane & 15;                // N within tile
  const int kk     = (lane >> 4) << 1;         // K-half base: 0 or 2

  const float* xs = x   + (size_t)sample * D;
  const float* w1 = w1s + (size_t)leaf * (D * H);
  const float* b1 = b1s + (size_t)leaf * H;
  const float* w2 = w2s + (size_t)leaf * (H * O);
  const float* b2 = b2s + (size_t)leaf * O;

  // ---- GEMM1: h(1x32) = x(1x1024) @ W1(1024x32) ----
  v8f c0 = {}; v8f c1 = {};
  for (int kb = 0; kb < D; kb += 4) {
    v2f a;  a.x  = xs[kb + kk];                  a.y  = xs[kb + kk + 1];
    v2f bA; bA.x = w1[(kb + kk) * H + nsub];     bA.y = w1[(kb + kk + 1) * H + nsub];
    v2f bB; bB.x = w1[(kb + kk) * H + 16 + nsub];bB.y = w1[(kb + kk + 1) * H + 16 + nsub];
    c0 = __builtin_amdgcn_wmma_f32_16x16x4_f32(false, a, false, bA, (short)0, c0, false, false);
    c1 = __builtin_amdgcn_wmma_f32_16x16x4_f32(false, a, false, bB, (short)0, c1, false, false);
  }
  // C VGPR0, lanes 0-15: c0[0] = h[lane], c1[0] = h[16+lane]
  float* hw = hsm + wid * H;
  if (lane < 16) { hw[lane] = c0[0]; hw[16 + lane] = c1[0]; }
  __syncthreads();

  // ---- broadcast-A fragments for GEMM2 with bias + ReLU fused ----
  v2f af[8];
#pragma unroll
  for (int kc = 0; kc < 8; ++kc) {
    const int hi = 4 * kc + kk;
    float h0 = hw[hi]     + b1[hi];
    float h1 = hw[hi + 1] + b1[hi + 1];
    af[kc].x = h0 > 0.f ? h0 : 0.f;
    af[kc].y = h1 > 0.f ? h1 : 0.f;
  }

  // ---- GEMM2: out(1x1024) = h(1x32) @ W2(32x1024), 2 N-tiles/iter ----
  float* os = out + (size_t)sample * O;
  for (int ob = 0; ob < O; ob += 32) {
    v8f ca = {}; v8f cb = {};
#pragma unroll
    for (int kc = 0; kc < 8; ++kc) {
      const int row0 = (4 * kc + kk) * O;
      const int row1 = row0 + O;
      v2f bA; bA.x = w2[row0 + ob + nsub];      bA.y = w2[row1 + ob + nsub];
      v2f bB; bB.x = w2[row0 + ob + 16 + nsub]; bB.y = w2[row1 + ob + 16 + nsub];
      ca = __builtin_amdgcn_wmma_f32_16x16x4_f32(false, af[kc], false, bA, (short)0, ca, false, false);
      cb = __builtin_amdgcn_wmma_f32_16x16x4_f32(false, af[kc], false, bB, (short)0, cb, false, false);
    }
    // lanes 0-15: ca[0] = out[ob+lane]; lanes 16-31: cb[0] = out[ob+lane]
    os[ob + lane] = (lane < 16 ? ca[0] : cb[0]) + b2[ob + lane];
  }
}

// ---------------------------------------------------------------------------
extern "C" void kernel_launch(void* const* d_in, const int* in_sizes, int n_in,
                              void* d_out, int out_size, void* d_ws, size_t ws_size,
                              hipStream_t stream) {
  const float* x   = (const float*)d_in[0];
  const float* nw  = (const float*)d_in[1];
  const float* nb  = (const float*)d_in[2];
  const float* w1s = (const float*)d_in[3];
  const float* b1s = (const float*)d_in[4];
  const float* w2s = (const float*)d_in[5];
  const float* b2s = (const float*)d_in[6];
  int* leaves = (int*)d_ws;                    // 4096 ints of scratch

  fff_descent<<<(BATCH * 32) / 256, 256, 0, stream>>>(x, nw, nb, leaves);
  fff_leaf_mlp<<<BATCH / 8, 256, 0, stream>>>(x, w1s, b1s, w2s, b2s, leaves,
                                              (float*)d_out);
}